// AddAtt_60395830116823
// MI455X (gfx1250) — compile-verified
//
#include <hip/hip_runtime.h>

// Problem constants (from reference)
#define BB 4
#define TK 512
#define TQ 512
#define KEYSZ 256
#define HH 128
#define VALSZ 256

typedef __attribute__((ext_vector_type(2))) float v2f;
typedef __attribute__((ext_vector_type(8))) float v8f;

// ---------------------------------------------------------------------------
// Generic batched f32 GEMM using V_WMMA_F32_16X16X4_F32.
// D[b] = A[b](MxK, row-major) * Bm[b](KxN, row-major) (+ bias[N] if non-null)
// One wave (32 lanes) per 16x16 output tile. M,N %16 == 0, K %4 == 0.
// ---------------------------------------------------------------------------
__global__ __launch_bounds__(128)
void wmma_gemm_f32(const float* __restrict__ A, const float* __restrict__ Bm,
                   const float* __restrict__ bias, float* __restrict__ D,
                   int M, int N, int K,
                   long long strideA, long long strideB, long long strideD) {
    const int wave = threadIdx.x >> 5;
    const int tile = blockIdx.x * 4 + wave;
    const int nTiles = N >> 4;
    const int numTiles = (M >> 4) * nTiles;
    if (tile >= numTiles) return;              // wave-uniform exit

    const int mBase = (tile / nTiles) << 4;
    const int nBase = (tile % nTiles) << 4;

    const int lane = threadIdx.x & 31;
    const int half = lane >> 4;                // 0: lanes 0-15, 1: lanes 16-31
    const int lo   = lane & 15;

    const float* Ab = A  + (size_t)blockIdx.y * strideA;
    const float* Bb = Bm + (size_t)blockIdx.y * strideB;
    float*       Db = D  + (size_t)blockIdx.y * strideD;

    v8f acc = {};
    for (int k0 = 0; k0 < K; k0 += 4) {
        // A fragment 16x4: VGPR r, half h -> K = r + 2h  (ISA 32-bit A layout)
        const float* ap = Ab + (size_t)(mBase + lo) * K + (k0 + 2 * half);
        v2f a;
        a.x = ap[0];
        a.y = ap[1];
        // B fragment 4x16: VGPR r, half h -> K = r + 2h, N = lo
        const float* bp = Bb + (size_t)(k0 + 2 * half) * N + (nBase + lo);
        v2f b;
        b.x = bp[0];
        b.y = bp[N];
        acc = __builtin_amdgcn_wmma_f32_16x16x4_f32(
            /*neg_a=*/false, a, /*neg_b=*/false, b,
            /*c_mod=*/(short)0, acc, /*reuse_a=*/false, /*reuse_b=*/false);
    }

    const float bval = bias ? bias[nBase + lo] : 0.0f;
    #pragma unroll
    for (int r = 0; r < 8; ++r) {
        // C/D layout: VGPR r, half h -> M = r + 8h, N = lo
        Db[(size_t)(mBase + r + 8 * half) * N + (nBase + lo)] = acc[r] + bval;
    }
}

// ---------------------------------------------------------------------------
// Fused scores + mask + softmax.
// Block: 256 threads handles (batch b, 4 consecutive k-rows), all TQ columns.
//   scores[kk][t] = sum_h w_v[h] * tanh(kproj[b,kk,h] + qproj[b,t,h]) + b_v
//   masked softmax over t (fill -1e6 where t >= valid_len), write attn.
// ---------------------------------------------------------------------------
#define KB 4
#define NT 256

__global__ __launch_bounds__(NT)
void scores_softmax(const float* __restrict__ kproj, const float* __restrict__ qproj,
                    const float* __restrict__ w_v, const float* __restrict__ b_v,
                    const int* __restrict__ valid_lens, float* __restrict__ attn) {
    __shared__ float kv[KB * HH];     // k-row tile
    __shared__ float wv[HH];
    __shared__ float srow[KB * TQ];   // score rows
    __shared__ float red[NT];

    const int tid = threadIdx.x;
    const int b   = blockIdx.y;
    const int kk0 = blockIdx.x * KB;

    // cooperative loads
    for (int i = tid; i < KB * HH; i += NT)
        kv[i] = kproj[((size_t)b * TK + kk0) * HH + i];
    if (tid < HH) wv[tid] = w_v[tid];
    __syncthreads();

    const float bv = b_v[0];

    // compute scores: each thread owns 2 t-values across all KB rows
    for (int ti = 0; ti < TQ / NT; ++ti) {
        const int t = tid + ti * NT;
        const float* qrow = qproj + ((size_t)b * TQ + t) * HH;
        float a0 = 0.f, a1 = 0.f, a2 = 0.f, a3 = 0.f;
        for (int h = 0; h < HH; ++h) {
            const float qh = qrow[h];
            const float w  = wv[h];
            a0 += w * tanhf(kv[0 * HH + h] + qh);
            a1 += w * tanhf(kv[1 * HH + h] + qh);
            a2 += w * tanhf(kv[2 * HH + h] + qh);
            a3 += w * tanhf(kv[3 * HH + h] + qh);
        }
        srow[0 * TQ + t] = a0 + bv;
        srow[1 * TQ + t] = a1 + bv;
        srow[2 * TQ + t] = a2 + bv;
        srow[3 * TQ + t] = a3 + bv;
    }
    __syncthreads();

    // masked softmax per row
    for (int r = 0; r < KB; ++r) {
        const int vlen = valid_lens[(size_t)b * TK + kk0 + r];

        float lmax = -3.402823e38f;
        for (int t = tid; t < TQ; t += NT) {
            float s = (t < vlen) ? srow[r * TQ + t] : -1.0e6f;
            srow[r * TQ + t] = s;
            lmax = fmaxf(lmax, s);
        }
        red[tid] = lmax;
        __syncthreads();
        for (int off = NT / 2; off > 0; off >>= 1) {
            if (tid < off) red[tid] = fmaxf(red[tid], red[tid + off]);
            __syncthreads();
        }
        const float m = red[0];
        __syncthreads();

        float lsum = 0.f;
        for (int t = tid; t < TQ; t += NT) {
            const float e = expf(srow[r * TQ + t] - m);
            srow[r * TQ + t] = e;
            lsum += e;
        }
        red[tid] = lsum;
        __syncthreads();
        for (int off = NT / 2; off > 0; off >>= 1) {
            if (tid < off) red[tid] += red[tid + off];
            __syncthreads();
        }
        const float inv = 1.0f / red[0];
        __syncthreads();

        float* arow = attn + ((size_t)b * TK + kk0 + r) * TQ;
        for (int t = tid; t < TQ; t += NT)
            arow[t] = srow[r * TQ + t] * inv;
        __syncthreads();
    }
}

// ---------------------------------------------------------------------------
extern "C" void kernel_launch(void* const* d_in, const int* in_sizes, int n_in,
                              void* d_out, int out_size, void* d_ws, size_t ws_size,
                              hipStream_t stream) {
    const float* key        = (const float*)d_in[0];  // [B,TK,KEYSZ]
    const float* que        = (const float*)d_in[1];  // [B,TQ,KEYSZ]
    const float* value      = (const float*)d_in[2];  // [B,TQ,VALSZ]
    const float* W_k        = (const float*)d_in[3];  // [KEYSZ,HH]
    const float* b_k        = (const float*)d_in[4];  // [HH]
    const float* W_q        = (const float*)d_in[5];  // [KEYSZ,HH]
    const float* b_q        = (const float*)d_in[6];  // [HH]
    const float* w_v        = (const float*)d_in[7];  // [HH]
    const float* b_v        = (const float*)d_in[8];  // scalar
    const int*   valid_lens = (const int*)d_in[9];    // [B,TK]
    float* out = (float*)d_out;                       // [B,TK,VALSZ]

    // workspace: kproj | qproj | attn  (6 MB total, f32)
    float* kproj = (float*)d_ws;
    float* qproj = kproj + (size_t)BB * TK * HH;
    float* attn  = qproj + (size_t)BB * TQ * HH;

    // 1) projections: (B*TK x KEYSZ) @ (KEYSZ x HH) + bias  (batches folded into M)
    {
        const int M = BB * TK, N = HH, K = KEYSZ;
        const int tiles = (M / 16) * (N / 16);
        dim3 grid((tiles + 3) / 4, 1);
        wmma_gemm_f32<<<grid, 128, 0, stream>>>(key, W_k, b_k, kproj, M, N, K, 0, 0, 0);
        wmma_gemm_f32<<<grid, 128, 0, stream>>>(que, W_q, b_q, qproj, M, N, K, 0, 0, 0);
    }

    // 2) fused scores + tanh + masked softmax -> attn
    {
        dim3 grid(TK / KB, BB);
        scores_softmax<<<grid, NT, 0, stream>>>(kproj, qproj, w_v, b_v, valid_lens, attn);
    }

    // 3) out[b] = attn[b] (TK x TQ) @ value[b] (TQ x VALSZ), batched over B
    {
        const int M = TK, N = VALSZ, K = TQ;
        const int tiles = (M / 16) * (N / 16);
        dim3 grid((tiles + 3) / 4, BB);
        wmma_gemm_f32<<<grid, 128, 0, stream>>>(attn, value, nullptr, out, M, N, K,
                                                (long long)TK * TQ,
                                                (long long)TQ * VALSZ,
                                                (long long)TK * VALSZ);
    }
}